// FocalLossAdaptive_42958262894789
// MI455X (gfx1250) — compile-verified
//
#include <hip/hip_runtime.h>
#include <hip/hip_bf16.h>
#include <stdint.h>

#define THREADS 256
// float4 elements per async stage. 1000 % 256 = 232 (> 224), so EVERY wave32
// issues exactly 4 global_load_async_to_lds_b128 per stage -> per-wave
// s_wait_asynccnt thresholds (8 = two stages in flight, 4 = one) are exact.
#define CHUNK 1000
#define NBUF  3   // depth-2 async pipeline: stages s+1, s+2 in flight while consuming s

// Branchless online-softmax update at float4 granularity: 5 exps / 4 elements
// (vs 2 exps/element naively) -> keeps v_exp_f32 pressure under the HBM roofline.
__device__ __forceinline__ void online_quad(const float4 v, float& m, float& s) {
  float vmax = fmaxf(fmaxf(v.x, v.y), fmaxf(v.z, v.w));
  float M  = fmaxf(m, vmax);
  float e0 = __expf(m - M);
  float ex = __expf(v.x - M);
  float ey = __expf(v.y - M);
  float ez = __expf(v.z - M);
  float ew = __expf(v.w - M);
  s = fmaf(s, e0, (ex + ey) + (ez + ew));
  m = M;
}

__device__ __forceinline__ void online_one(float v, float& m, float& s) {
  float M = fmaxf(m, v);
  s = s * __expf(m - M) + __expf(v - M);
  m = M;
}

__global__ __launch_bounds__(THREADS) void focal_rows_kernel(
    const float* __restrict__ input, const int* __restrict__ target,
    float* __restrict__ row_loss, int C) {
  __shared__ float4 buf[NBUF][CHUNK];   // 48 KB triple buffer
  __shared__ float rm[8], rs[8];

  const int row = blockIdx.x;
  const float* grow = input + (size_t)row * (size_t)C;
  const float4* grow4 = reinterpret_cast<const float4*>(grow);
  const int f4n  = C >> 2;
  const int full = f4n / CHUNK;

  float xt = 0.0f;
  if (threadIdx.x == 0) xt = grow[target[row]];   // target logit (scalarized)

  float m = -3.402823466e38f, s = 0.0f;

  if (full > 0) {
    // prologue: issue stages 0 and 1
    for (int j = threadIdx.x; j < CHUNK; j += THREADS) {
      uint32_t loff = (uint32_t)(uintptr_t)(&buf[0][j]);  // low 32 bits == LDS offset
      asm volatile("global_load_async_to_lds_b128 %0, %1, off th:TH_LOAD_NT"
                   :: "v"(loff), "v"(grow4 + j) : "memory");
    }
    if (full > 1) {
      const float4* gp = grow4 + CHUNK;
      for (int j = threadIdx.x; j < CHUNK; j += THREADS) {
        uint32_t loff = (uint32_t)(uintptr_t)(&buf[1][j]);
        asm volatile("global_load_async_to_lds_b128 %0, %1, off th:TH_LOAD_NT"
                     :: "v"(loff), "v"(gp + j) : "memory");
      }
    }
    for (int st = 0; st < full; ++st) {
      const int nb = st + 2;
      if (nb < full) {
        const float4* gp = grow4 + nb * CHUNK;
        float4* lb = buf[nb % NBUF];
        for (int j = threadIdx.x; j < CHUNK; j += THREADS) {
          uint32_t loff = (uint32_t)(uintptr_t)(lb + j);
          asm volatile("global_load_async_to_lds_b128 %0, %1, off th:TH_LOAD_NT"
                       :: "v"(loff), "v"(gp + j) : "memory");
        }
        asm volatile("s_wait_asynccnt 8" ::: "memory");   // stage st complete; st+1,st+2 in flight
      } else if (st + 1 < full) {
        asm volatile("s_wait_asynccnt 4" ::: "memory");   // stage st complete; st+1 in flight
      } else {
        asm volatile("s_wait_asynccnt 0" ::: "memory");   // last stage complete
      }
      // consume stage st from LDS (each thread reads exactly what it loaded)
      const float4* lb = buf[st % NBUF];
      for (int j = threadIdx.x; j < CHUNK; j += THREADS)
        online_quad(lb[j], m, s);
    }
  }
  // float4 remainder (direct global loads) — none for C = 32000
  for (int j = full * CHUNK + threadIdx.x; j < f4n; j += THREADS)
    online_quad(grow4[j], m, s);
  // scalar remainder — none for C % 4 == 0
  for (int e = (f4n << 2) + threadIdx.x; e < C; e += THREADS)
    online_one(grow[e], m, s);

  // wave32 combine of (m, s)
  #pragma unroll
  for (int off = 16; off > 0; off >>= 1) {
    float mo = __shfl_xor(m, off, 32);
    float so = __shfl_xor(s, off, 32);
    float M = fmaxf(m, mo);
    s = s * __expf(m - M) + so * __expf(mo - M);
    m = M;
  }
  const int lane = threadIdx.x & 31;
  const int wid  = threadIdx.x >> 5;
  if (lane == 0) { rm[wid] = m; rs[wid] = s; }
  __syncthreads();
  if (threadIdx.x == 0) {
    float M = rm[0], S = rs[0];
    #pragma unroll
    for (int w = 1; w < 8; ++w) {
      float m2 = rm[w], s2 = rs[w];
      float Mn = fmaxf(M, m2);
      S = S * __expf(M - Mn) + s2 * __expf(m2 - Mn);
      M = Mn;
    }
    float lse   = M + __logf(S);
    float logpt = xt - lse;
    float pt    = __expf(logpt);
    float u  = 1.0f - pt;
    float u3 = u * u * u;
    float w  = (pt < 0.2f) ? u3 * u * u : u3;   // gamma: 5 if pt<0.2 else 3
    row_loss[row] = -w * logpt;
  }
}

__global__ __launch_bounds__(THREADS) void mean_kernel(
    const float* __restrict__ x, float* __restrict__ out, int n) {
  __shared__ float red[8];
  float acc = 0.0f;
  for (int i = threadIdx.x; i < n; i += THREADS) acc += x[i];
  #pragma unroll
  for (int off = 16; off > 0; off >>= 1) acc += __shfl_xor(acc, off, 32);
  if ((threadIdx.x & 31) == 0) red[threadIdx.x >> 5] = acc;
  __syncthreads();
  if (threadIdx.x == 0) {
    float v = 0.0f;
    #pragma unroll
    for (int w = 0; w < 8; ++w) v += red[w];
    out[0] = v / (float)n;
  }
}

extern "C" void kernel_launch(void* const* d_in, const int* in_sizes, int n_in,
                              void* d_out, int out_size, void* d_ws, size_t ws_size,
                              hipStream_t stream) {
  const float* input  = (const float*)d_in[0];
  const int*   target = (const int*)d_in[1];
  const int N = in_sizes[1];
  const int C = (int)((long long)in_sizes[0] / (long long)N);
  float* row_loss = (float*)d_ws;

  focal_rows_kernel<<<N, THREADS, 0, stream>>>(input, target, row_loss, C);
  mean_kernel<<<1, THREADS, 0, stream>>>(row_loss, (float*)d_out, N);
}